// MambaModel_26869315404120
// MI455X (gfx1250) — compile-verified
//
#include <hip/hip_runtime.h>
#include <hip/hip_bf16.h>
#include <cstdint>

// ---------------- Model constants (match reference) ----------------
#define B_SZ    2
#define L_SZ    2048
#define VOCAB   50257
#define D_MODEL 768
#define N_LAYERS 12
#define D_STATE 16
#define D_CONV  4
#define D_INNER 1536
#define DT_RANK 48
#define XDB_COLS (DT_RANK + 2*D_STATE)   // 80
#define EPS_RMS 1e-5f
#define M_ROWS  (B_SZ*L_SZ)              // 4096
#define NCHUNK  32
#define CLEN    (L_SZ / NCHUNK)          // 64

typedef float v2f __attribute__((ext_vector_type(2)));
typedef float v8f __attribute__((ext_vector_type(8)));

__device__ __forceinline__ float silu_f(float x)     { return x / (1.f + __expf(-x)); }
__device__ __forceinline__ float softplus_f(float x) { return (x > 20.f) ? x : log1pf(__expf(x)); }

// ---------------- fp32 WMMA NT-GEMM: C[M,N] = A[M,K] * W[N,K]^T ----------------
// Wave tile 32(M) x 64(N): 2 A fragments, 4 shared B fragments, 8 WMMAs / k-step.
// Block = 8 waves as 4(M) x 2(N) -> 128x128 block tile.
// OOB columns: row pointer clamped, loads unconditional (no EXEC juggling);
// garbage accumulates only into columns masked at the store.
// EPI: 0 = store, 1 = softplus(acc + bias[n]), 2 = C += acc (residual)
template<int EPI>
__global__ __launch_bounds__(256) void wmma_gemm_nt(
    const float* __restrict__ A, const float* __restrict__ Bw,
    float* __restrict__ C, const float* __restrict__ bias,
    int M, int N, int K, int lda, int ldb, int ldc)
{
    const int lane = threadIdx.x & 31;
    const int wave = threadIdx.x >> 5;
    const int wm = wave & 3;
    const int wn = wave >> 2;
    const int rowBase = blockIdx.y * 128 + wm * 32;
    const int colBase = blockIdx.x * 128 + wn * 64;
    if (rowBase >= M) return;

    const int m  = lane & 15;
    const int ks = lane >> 4;

    int r0 = rowBase + m;      if (r0 >= M) r0 = M - 1;
    int r1 = rowBase + 16 + m; if (r1 >= M) r1 = M - 1;
    const float* arow0 = A + (size_t)r0 * lda + 2 * ks;
    const float* arow1 = A + (size_t)r1 * lda + 2 * ks;

    const float* brow[4];
#pragma unroll
    for (int j = 0; j < 4; ++j) {
        int n = colBase + j * 16 + m;
        if (n >= N) n = N - 1;                  // clamp: safe unconditional loads
        brow[j] = Bw + (size_t)n * ldb + 2 * ks;
    }

    v8f acc0[4] = {};
    v8f acc1[4] = {};

    for (int k = 0; k < K; k += 4) {
        v2f a0 = *(const v2f*)(arow0 + k);
        v2f a1 = *(const v2f*)(arow1 + k);
#pragma unroll
        for (int j = 0; j < 4; ++j) {
            v2f b = *(const v2f*)(brow[j] + k);
            acc0[j] = __builtin_amdgcn_wmma_f32_16x16x4_f32(
                false, a0, false, b, (short)0, acc0[j], false, false);
            acc1[j] = __builtin_amdgcn_wmma_f32_16x16x4_f32(
                false, a1, false, b, (short)0, acc1[j], false, false);
        }
    }

#pragma unroll
    for (int j = 0; j < 4; ++j) {
        int n = colBase + j * 16 + m;
        if (n >= N) continue;
#pragma unroll
        for (int v = 0; v < 8; ++v) {
            int r = rowBase + v + 8 * ks;
            if (r < M) {
                size_t idx = (size_t)r * ldc + n;
                float val = acc0[j][v];
                if (EPI == 0)      C[idx] = val;
                else if (EPI == 1) C[idx] = softplus_f(val + bias[n]);
                else               C[idx] += val;
            }
            int r2 = rowBase + 16 + v + 8 * ks;
            if (r2 < M) {
                size_t idx = (size_t)r2 * ldc + n;
                float val = acc1[j][v];
                if (EPI == 0)      C[idx] = val;
                else if (EPI == 1) C[idx] = softplus_f(val + bias[n]);
                else               C[idx] += val;
            }
        }
    }
}

// ---------------- Embedding gather ----------------
__global__ void embed_kernel(const int* __restrict__ ids, const float* __restrict__ emb,
                             float* __restrict__ x, int total)
{
    int i = blockIdx.x * blockDim.x + threadIdx.x;
    if (i >= total) return;
    int r = i / D_MODEL, c = i % D_MODEL;
    x[i] = emb[(size_t)ids[r] * D_MODEL + c];
}

// ---------------- RMSNorm (one block per token row) ----------------
__global__ __launch_bounds__(256) void rmsnorm_kernel(const float* __restrict__ x,
                                                      const float* __restrict__ w,
                                                      float* __restrict__ out)
{
    __shared__ float red[256];
    const float* xr = x + (size_t)blockIdx.x * D_MODEL;
    float s = 0.f;
    for (int c = threadIdx.x; c < D_MODEL; c += 256) { float v = xr[c]; s += v * v; }
    red[threadIdx.x] = s;
    __syncthreads();
    for (int off = 128; off > 0; off >>= 1) {
        if (threadIdx.x < off) red[threadIdx.x] += red[threadIdx.x + off];
        __syncthreads();
    }
    float scale = rsqrtf(red[0] / (float)D_MODEL + EPS_RMS);
    for (int c = threadIdx.x; c < D_MODEL; c += 256)
        out[(size_t)blockIdx.x * D_MODEL + c] = xr[c] * scale * w[c];
}

// ---------------- Depthwise causal conv (k=4) + SiLU ----------------
__global__ void conv_silu_kernel(const float* __restrict__ xz, const float* __restrict__ cw,
                                 const float* __restrict__ cb, float* __restrict__ xp, int total)
{
    int i = blockIdx.x * blockDim.x + threadIdx.x;
    if (i >= total) return;
    int d  = i % D_INNER;
    int bt = i / D_INNER;
    int t  = bt % L_SZ;
    float s = cb[d];
#pragma unroll
    for (int j = 0; j < D_CONV; ++j) {
        int tt = t - (D_CONV - 1) + j;
        if (tt >= 0)
            s += cw[d * D_CONV + j] * xz[(size_t)(bt - (D_CONV - 1) + j) * (2 * D_INNER) + d];
    }
    xp[i] = silu_f(s);
}

// ---------------- Chunked parallel selective scan ----------------
// h_t = exp(a*dt_t)*h_{t-1} + dt_t*B_t*u_t  is an affine scan; chunk decay
// collapses to exp(a * sum(dt)).  Phase 1: per-chunk partials from zero state.
// Phase 2: sequential prefix over 32 chunk summaries per (b,d).
// Phase 3: within-chunk replay from composed initial state, emit y.

// Phase 1: thread per (b, chunk, d)
__global__ __launch_bounds__(256) void scan_phase1(const float* __restrict__ xp,
                                                   const float* __restrict__ dtb,
                                                   const float* __restrict__ xdb,
                                                   const float* __restrict__ A_log,
                                                   float* __restrict__ sdt,
                                                   float* __restrict__ pend)
{
    int gid = blockIdx.x * 256 + threadIdx.x;
    if (gid >= B_SZ * NCHUNK * D_INNER) return;
    int b = gid / (NCHUNK * D_INNER);
    int rem = gid % (NCHUNK * D_INNER);
    int c = rem / D_INNER;
    int d = rem % D_INNER;

    float a[D_STATE], h[D_STATE];
#pragma unroll
    for (int s = 0; s < D_STATE; ++s) { a[s] = -__expf(A_log[d * D_STATE + s]); h[s] = 0.f; }
    float S = 0.f;

    for (int i = 0; i < CLEN; ++i) {
        size_t bt = (size_t)b * L_SZ + c * CLEN + i;
        float dt = dtb[bt * D_INNER + d];
        float u  = xp [bt * D_INNER + d];
        const float* bc = xdb + bt * XDB_COLS;
        S += dt;
#pragma unroll
        for (int s = 0; s < D_STATE; ++s)
            h[s] = __expf(dt * a[s]) * h[s] + dt * bc[DT_RANK + s] * u;
    }
    sdt[gid] = S;
#pragma unroll
    for (int s = 0; s < D_STATE; ++s) pend[(size_t)gid * D_STATE + s] = h[s];
}

// Phase 2: thread per (b, d); 32-step prefix over chunk summaries
__global__ __launch_bounds__(256) void scan_phase2(const float* __restrict__ A_log,
                                                   const float* __restrict__ sdt,
                                                   const float* __restrict__ pend,
                                                   float* __restrict__ hst)
{
    int gid = blockIdx.x * 256 + threadIdx.x;
    if (gid >= B_SZ * D_INNER) return;
    int b = gid / D_INNER, d = gid % D_INNER;

    float a[D_STATE], h[D_STATE];
#pragma unroll
    for (int s = 0; s < D_STATE; ++s) { a[s] = -__expf(A_log[d * D_STATE + s]); h[s] = 0.f; }

    for (int c = 0; c < NCHUNK; ++c) {
        size_t idx = ((size_t)b * NCHUNK + c) * D_INNER + d;
#pragma unroll
        for (int s = 0; s < D_STATE; ++s) hst[idx * D_STATE + s] = h[s];  // state at chunk start
        float dec = sdt[idx];
#pragma unroll
        for (int s = 0; s < D_STATE; ++s)
            h[s] = __expf(a[s] * dec) * h[s] + pend[idx * D_STATE + s];
    }
}

// Phase 3: thread per (b, chunk, d); replay chunk from composed state, emit y
__global__ __launch_bounds__(256) void scan_phase3(const float* __restrict__ xp,
                                                   const float* __restrict__ dtb,
                                                   const float* __restrict__ xdb,
                                                   const float* __restrict__ A_log,
                                                   const float* __restrict__ Dp,
                                                   const float* __restrict__ hst,
                                                   float* __restrict__ y)
{
    int gid = blockIdx.x * 256 + threadIdx.x;
    if (gid >= B_SZ * NCHUNK * D_INNER) return;
    int b = gid / (NCHUNK * D_INNER);
    int rem = gid % (NCHUNK * D_INNER);
    int c = rem / D_INNER;
    int d = rem % D_INNER;

    float a[D_STATE], h[D_STATE];
#pragma unroll
    for (int s = 0; s < D_STATE; ++s) {
        a[s] = -__expf(A_log[d * D_STATE + s]);
        h[s] = hst[(size_t)gid * D_STATE + s];
    }
    float Dv = Dp[d];

    for (int i = 0; i < CLEN; ++i) {
        size_t bt = (size_t)b * L_SZ + c * CLEN + i;
        float dt = dtb[bt * D_INNER + d];
        float u  = xp [bt * D_INNER + d];
        const float* bc = xdb + bt * XDB_COLS;
        float yv = 0.f;
#pragma unroll
        for (int s = 0; s < D_STATE; ++s) {
            h[s] = __expf(dt * a[s]) * h[s] + dt * bc[DT_RANK + s] * u;
            yv  += h[s] * bc[DT_RANK + D_STATE + s];
        }
        y[bt * D_INNER + d] = yv + u * Dv;
    }
}

// ---------------- Gate: y *= silu(z) ----------------
__global__ void gate_kernel(const float* __restrict__ xz, float* __restrict__ y, int total)
{
    int i = blockIdx.x * blockDim.x + threadIdx.x;
    if (i >= total) return;
    int d  = i % D_INNER;
    int bt = i / D_INNER;
    float z = xz[(size_t)bt * (2 * D_INNER) + D_INNER + d];
    y[i] = y[i] * silu_f(z);
}

// ---------------- Host orchestration ----------------
extern "C" void kernel_launch(void* const* d_in, const int* in_sizes, int n_in,
                              void* d_out, int out_size, void* d_ws, size_t ws_size,
                              hipStream_t stream)
{
    const int*   ids      = (const int*)  d_in[0];
    const float* emb      = (const float*)d_in[1];
    const float* norm_w   = (const float*)d_in[2];
    const float* W_in     = (const float*)d_in[3];
    const float* conv_w   = (const float*)d_in[4];
    const float* conv_b   = (const float*)d_in[5];
    const float* W_x      = (const float*)d_in[6];
    const float* W_dt     = (const float*)d_in[7];
    const float* b_dt     = (const float*)d_in[8];
    const float* A_log    = (const float*)d_in[9];
    const float* Dp       = (const float*)d_in[10];
    const float* W_out    = (const float*)d_in[11];
    const float* norm_f_w = (const float*)d_in[12];
    float* out = (float*)d_out;

    // Workspace layout (floats)
    float* ws = (float*)d_ws;
    const size_t nX    = (size_t)M_ROWS * D_MODEL;
    const size_t nXZ   = (size_t)M_ROWS * 2 * D_INNER;
    const size_t nI    = (size_t)M_ROWS * D_INNER;
    const size_t nXDB  = (size_t)M_ROWS * XDB_COLS;
    const size_t nSDT  = (size_t)B_SZ * NCHUNK * D_INNER;
    const size_t nPEND = nSDT * D_STATE;
    float* x    = ws;
    float* xn   = x    + nX;
    float* xz   = xn   + nX;
    float* xp   = xz   + nXZ;
    float* dtb  = xp   + nI;
    float* xdb  = dtb  + nI;
    float* y    = xdb  + nXDB;
    float* sdt  = y    + nI;
    float* pend = sdt  + nSDT;
    float* hst  = pend + nPEND;

    dim3 blk(256);
    const int gy = (M_ROWS + 127) / 128;   // 32

    embed_kernel<<<(int)((nX + 255) / 256), blk, 0, stream>>>(ids, emb, x, (int)nX);

    for (int l = 0; l < N_LAYERS; ++l) {
        const float* nw  = norm_w + (size_t)l * D_MODEL;
        const float* Wi  = W_in   + (size_t)l * 2 * D_INNER * D_MODEL;
        const float* cw  = conv_w + (size_t)l * D_INNER * D_CONV;
        const float* cb  = conv_b + (size_t)l * D_INNER;
        const float* Wx  = W_x    + (size_t)l * XDB_COLS * D_INNER;
        const float* Wdt = W_dt   + (size_t)l * D_INNER * DT_RANK;
        const float* bdt = b_dt   + (size_t)l * D_INNER;
        const float* Al  = A_log  + (size_t)l * D_INNER * D_STATE;
        const float* Dpl = Dp     + (size_t)l * D_INNER;
        const float* Wo  = W_out  + (size_t)l * D_MODEL * D_INNER;

        rmsnorm_kernel<<<M_ROWS, blk, 0, stream>>>(x, nw, xn);

        // xz = xn @ W_in^T   (4096 x 3072 x 768)
        wmma_gemm_nt<0><<<dim3((2 * D_INNER + 127) / 128, gy), blk, 0, stream>>>(
            xn, Wi, xz, nullptr, M_ROWS, 2 * D_INNER, D_MODEL, D_MODEL, D_MODEL, 2 * D_INNER);

        conv_silu_kernel<<<(int)((nI + 255) / 256), blk, 0, stream>>>(xz, cw, cb, xp, (int)nI);

        // xdb = xp @ W_x^T   (4096 x 80 x 1536)
        wmma_gemm_nt<0><<<dim3((XDB_COLS + 127) / 128, gy), blk, 0, stream>>>(
            xp, Wx, xdb, nullptr, M_ROWS, XDB_COLS, D_INNER, D_INNER, D_INNER, XDB_COLS);

        // dt = softplus(xdb[:, :48] @ W_dt^T + b_dt)   (4096 x 1536 x 48)
        wmma_gemm_nt<1><<<dim3((D_INNER + 127) / 128, gy), blk, 0, stream>>>(
            xdb, Wdt, dtb, bdt, M_ROWS, D_INNER, DT_RANK, XDB_COLS, DT_RANK, D_INNER);

        // chunked selective scan
        scan_phase1<<<(int)((nSDT + 255) / 256), blk, 0, stream>>>(xp, dtb, xdb, Al, sdt, pend);
        scan_phase2<<<(B_SZ * D_INNER + 255) / 256, blk, 0, stream>>>(Al, sdt, pend, hst);
        scan_phase3<<<(int)((nSDT + 255) / 256), blk, 0, stream>>>(xp, dtb, xdb, Al, Dpl, hst, y);

        gate_kernel<<<(int)((nI + 255) / 256), blk, 0, stream>>>(xz, y, (int)nI);

        // x += y_gated @ W_out^T   (4096 x 768 x 1536), residual fused (EPI=2)
        wmma_gemm_nt<2><<<dim3((D_MODEL + 127) / 128, gy), blk, 0, stream>>>(
            y, Wo, x, nullptr, M_ROWS, D_MODEL, D_INNER, D_INNER, D_INNER, D_MODEL);
    }

    rmsnorm_kernel<<<M_ROWS, blk, 0, stream>>>(x, norm_f_w, xn);

    // logits = xn @ emb^T   (4096 x 50257 x 768)
    wmma_gemm_nt<0><<<dim3((VOCAB + 127) / 128, gy), blk, 0, stream>>>(
        xn, emb, out, nullptr, M_ROWS, VOCAB, D_MODEL, D_MODEL, D_MODEL, VOCAB);
}